// NeuronMemory_38491496907061
// MI455X (gfx1250) — compile-verified
//
#include <hip/hip_runtime.h>
#include <math.h>

// Problem constants (match reference)
#define B_   4
#define S_   2048
#define D_   1024
#define R_   128
#define NC_  64
#define NK_  32768
#define SD_  64
#define KK_  8

typedef float v2f __attribute__((ext_vector_type(2)));
typedef float v8f __attribute__((ext_vector_type(8)));
typedef int   v4i __attribute__((ext_vector_type(4)));
typedef __attribute__((address_space(1))) v4i gv4i; // global
typedef __attribute__((address_space(3))) v4i lv4i; // LDS

#if __has_builtin(__builtin_amdgcn_global_load_async_to_lds_b128) && \
    __has_builtin(__builtin_amdgcn_s_wait_asynccnt)
#define USE_ASYNC_LDS 1
#else
#define USE_ASYNC_LDS 0
#endif

// ---------------------------------------------------------------------------
// WMMA fragment loaders for V_WMMA_F32_16X16X4_F32 (wave32).
// A (16x4, MxK):  lanes 0-15 hold M=lane, K=0..1 in v0,v1; lanes 16-31 hold
//                 M=lane-16, K=2..3.
// B (4x16, KxN):  lanes 0-15 hold N=lane, K=0..1; lanes 16-31 N=lane-16, K=2..3.
// C/D (16x16):    VGPR i <-> M = (lane>=16 ? 8 : 0)+i, N = lane&15.
// ---------------------------------------------------------------------------
__device__ __forceinline__ v2f load_a_rm(const float* A, int lda, int row_base,
                                         int k0, int lane) {
  int half = lane >> 4, l = lane & 15;
  const float* p = A + (long long)(row_base + l) * lda + (k0 + half * 2);
  v2f a; a.x = p[0]; a.y = p[1];
  return a;
}
// B stored row-major [K,N]
__device__ __forceinline__ v2f load_b_kn(const float* Bm, int ldb, int k0,
                                         int col_base, int lane) {
  int half = lane >> 4, l = lane & 15;
  int k = k0 + half * 2;
  v2f b;
  b.x = Bm[(long long)(k    ) * ldb + col_base + l];
  b.y = Bm[(long long)(k + 1) * ldb + col_base + l];
  return b;
}
// B = W^T where W stored row-major [N,K]
__device__ __forceinline__ v2f load_b_nk(const float* W, int ldw, int k0,
                                         int col_base, int lane) {
  int half = lane >> 4, l = lane & 15;
  const float* p = W + (long long)(col_base + l) * ldw + (k0 + half * 2);
  v2f b; b.x = p[0]; b.y = p[1];
  return b;
}
__device__ __forceinline__ void store_d(float* Dp, int ldd, int row_base,
                                        int col_base, int lane, v8f c) {
  int half = lane >> 4, l = lane & 15;
#pragma unroll
  for (int i = 0; i < 8; ++i)
    Dp[(long long)(row_base + half * 8 + i) * ldd + col_base + l] = c[i];
}

// ---------------------------------------------------------------------------
// Generic batched WMMA f32 GEMM:  D[M,N] = A[M,K] @ (BT ? W[N,K]^T : B[K,N])
// One wave (32 threads) per 16x16 output tile.
// ---------------------------------------------------------------------------
template <bool BT>
__global__ void wmma_gemm_f32(const float* __restrict__ A,
                              const float* __restrict__ Bm,
                              float* __restrict__ Dp, int K, int N, int tiles_n,
                              long long strideA, long long strideB,
                              long long strideD) {
  const float* Ab = A  + (long long)blockIdx.y * strideA;
  const float* Bb = Bm + (long long)blockIdx.y * strideB;
  float*       Db = Dp + (long long)blockIdx.y * strideD;
  int tn = blockIdx.x % tiles_n;
  int tm = blockIdx.x / tiles_n;
  int lane = threadIdx.x;
  v8f acc = {};
  for (int k0 = 0; k0 < K; k0 += 4) {
    v2f a = load_a_rm(Ab, K, tm * 16, k0, lane);
    v2f b = BT ? load_b_nk(Bb, K, k0, tn * 16, lane)
               : load_b_kn(Bb, N, k0, tn * 16, lane);
    acc = __builtin_amdgcn_wmma_f32_16x16x4_f32(false, a, false, b, (short)0,
                                                acc, false, false);
  }
  store_d(Db, N, tm * 16, tn * 16, lane, acc);
}

__global__ void zero_kernel(float* p, int n) {
  int i = blockIdx.x * blockDim.x + threadIdx.x;
  if (i < n) p[i] = 0.f;
}

// ---------------------------------------------------------------------------
// Sequential SSM scan: h_t = h_{t-1} @ A + xB_t ; then h_proj = h @ W_imp^T
// ---------------------------------------------------------------------------
__global__ void scan_kernel(const float* __restrict__ xb,   // [B,S,SD]
                            const float* __restrict__ Amat, // [SD,SD]
                            const float* __restrict__ Wimp, // [D,SD]
                            float* __restrict__ h_proj) {   // [B,D]
  __shared__ float As[SD_ * SD_];
  __shared__ float h[B_ * SD_];
  int t = threadIdx.x; // 256
  for (int i = t; i < SD_ * SD_; i += 256) As[i] = Amat[i];
  h[t] = 0.f; // 256 == B_*SD_
  __syncthreads();
  int b = t >> 6, e = t & 63;
  for (int s = 0; s < S_; ++s) {
    float acc = xb[((long long)b * S_ + s) * SD_ + e];
#pragma unroll 8
    for (int j = 0; j < SD_; ++j) acc += h[b * SD_ + j] * As[j * SD_ + e];
    __syncthreads();
    h[t] = acc;
    __syncthreads();
  }
  for (int i = t; i < B_ * D_; i += 256) {
    int bb = i >> 10, d = i & (D_ - 1);
    float acc = 0.f;
#pragma unroll 8
    for (int ee = 0; ee < SD_; ++ee) acc += h[bb * SD_ + ee] * Wimp[d * SD_ + ee];
    h_proj[i] = acc;
  }
}

// ---------------------------------------------------------------------------
// importance logits: logit[b,s] = dot(x[b,s,:], h_proj[b,:])  (one wave/token)
// ---------------------------------------------------------------------------
__global__ void imp_logits_kernel(const float* __restrict__ x,
                                  const float* __restrict__ h_proj,
                                  float* __restrict__ logits) {
  int token = blockIdx.x; // B*S
  int b = token >> 11;
  int lane = threadIdx.x; // 32
  const float* xr = x + (long long)token * D_;
  const float* hp = h_proj + b * D_;
  float acc = 0.f;
  for (int i = lane; i < D_; i += 32) acc += xr[i] * hp[i];
#pragma unroll
  for (int o = 16; o > 0; o >>= 1) acc += __shfl_xor(acc, o, 32);
  if (lane == 0) logits[token] = acc;
}

// ---------------------------------------------------------------------------
// per-batch softmax over S
// ---------------------------------------------------------------------------
__global__ void imp_softmax_kernel(const float* __restrict__ logits,
                                   float* __restrict__ importance) {
  int b = blockIdx.x, t = threadIdx.x;
  __shared__ float red[256];
  const float* L = logits + b * S_;
  float m = -1e30f;
  for (int i = t; i < S_; i += 256) m = fmaxf(m, L[i]);
  red[t] = m; __syncthreads();
  for (int o = 128; o > 0; o >>= 1) {
    if (t < o) red[t] = fmaxf(red[t], red[t + o]);
    __syncthreads();
  }
  m = red[0]; __syncthreads();
  float s = 0.f;
  for (int i = t; i < S_; i += 256) s += __expf(L[i] - m);
  red[t] = s; __syncthreads();
  for (int o = 128; o > 0; o >>= 1) {
    if (t < o) red[t] += red[t + o];
    __syncthreads();
  }
  float inv = 1.f / red[0];
  for (int i = t; i < S_; i += 256)
    importance[b * S_ + i] = __expf(L[i] - m) * inv;
}

// ---------------------------------------------------------------------------
// token_pref softmax (over NC=64) weighted by importance -> neuron_w_raw
// ---------------------------------------------------------------------------
__global__ void tp_accum_kernel(const float* __restrict__ tp,  // [B*S, NC]
                                const float* __restrict__ importance,
                                float* __restrict__ nw_raw) {  // [B, NC]
  int token = blockIdx.x * 256 + threadIdx.x;
  int b = token >> 11;
  __shared__ float part[NC_];
  if (threadIdx.x < NC_) part[threadIdx.x] = 0.f;
  __syncthreads();
  const float* v = tp + (long long)token * NC_;
  float m = -1e30f;
#pragma unroll 8
  for (int n = 0; n < NC_; ++n) m = fmaxf(m, v[n]);
  float s = 0.f;
#pragma unroll 8
  for (int n = 0; n < NC_; ++n) s += __expf(v[n] - m);
  float w = importance[token] / s;
#pragma unroll 8
  for (int n = 0; n < NC_; ++n) atomicAdd(&part[n], __expf(v[n] - m) * w);
  __syncthreads();
  if (threadIdx.x < NC_)
    atomicAdd(&nw_raw[b * NC_ + threadIdx.x], part[threadIdx.x]);
}

__global__ void nw_norm_kernel(const float* __restrict__ nw_raw,
                               float* __restrict__ nw_ws,
                               float* __restrict__ nw_out) {
  int b = blockIdx.x, t = threadIdx.x; // 64
  __shared__ float red[NC_];
  float v = nw_raw[b * NC_ + t];
  red[t] = v; __syncthreads();
  for (int o = 32; o > 0; o >>= 1) {
    if (t < o) red[t] += red[t + o];
    __syncthreads();
  }
  float r = v / (red[0] + 1e-8f);
  nw_ws[b * NC_ + t] = r;
  nw_out[b * NC_ + t] = r;
}

// ---------------------------------------------------------------------------
// shared_compress[b,d,r] = sum_n nw[b,n] * compress[n,d,r]
// ---------------------------------------------------------------------------
__global__ void sc_kernel(const float* __restrict__ compress, // [NC,D,R]
                          const float* __restrict__ nw,       // [B,NC]
                          float* __restrict__ sc) {           // [B,D,R]
  const int DR = D_ * R_;
  int idx = blockIdx.x * 256 + threadIdx.x; // 0..DR-1
  __shared__ float w[B_ * NC_];
  w[threadIdx.x] = nw[threadIdx.x]; // 256 == B_*NC_
  __syncthreads();
  float a0 = 0.f, a1 = 0.f, a2 = 0.f, a3 = 0.f;
#pragma unroll 4
  for (int n = 0; n < NC_; ++n) {
    float c = compress[(long long)n * DR + idx];
    a0 += w[n] * c;
    a1 += w[NC_ + n] * c;
    a2 += w[2 * NC_ + n] * c;
    a3 += w[3 * NC_ + n] * c;
  }
  sc[idx] = a0;
  sc[DR + idx] = a1;
  sc[2 * DR + idx] = a2;
  sc[3 * DR + idx] = a3;
}

// ---------------------------------------------------------------------------
// Fused retrieval with shared, async double-buffered K tiles.
//
// Block = 128 threads = 4 waves; each wave owns a 16-token sub-tile and sees
// ALL 32768 knowledge rows (no cross-wave merge needed).  The 16x128 K tile
// (8 KB) is staged into LDS once per block and consumed by all 4 waves ->
// 4x less L2 traffic than per-wave streaming.  Staging uses
// GLOBAL_LOAD_ASYNC_TO_LDS_B128 + s_wait_asynccnt double-buffering when the
// toolchain exposes the builtin, else a synchronous float4 copy.
//
// LDS K row pitch = 132 dwords: keeps 16 B alignment and spreads the 16
// B-fragment ds_load_b64 lanes across banks (132 mod 64 = 4).
// ---------------------------------------------------------------------------
#define KT_LD   132
#define WAVES_  4
#define TOKPB_  (WAVES_ * 16)   // 64 tokens per block

__device__ __forceinline__ void stage_k_tile(const float* __restrict__ Kmat,
                                             float* __restrict__ ktbuf,
                                             int n_base, int tid) {
  // 16 rows x 128 floats = 512 chunks of 16 B; 128 threads x 4 chunks.
#pragma unroll
  for (int c = tid; c < 512; c += 128) {
    int row = c >> 5, col = c & 31; // col in 16B units
    const float* g = Kmat + (long long)(n_base + row) * R_ + col * 4;
    float* l = ktbuf + row * KT_LD + col * 4;
#if USE_ASYNC_LDS
    __builtin_amdgcn_global_load_async_to_lds_b128(
        (gv4i*)(float*)(void*)const_cast<float*>(g),
        (lv4i*)l, 0, 0);
#else
    *(float4*)l = *(const float4*)g;
#endif
  }
}

__global__ void retrieve_kernel(const float* __restrict__ Q,    // [B,S,R]
                                const float* __restrict__ Kmat, // [NK,R]
                                const float* __restrict__ V,    // [NK,D]
                                float* __restrict__ out,        // [B,S,D]
                                int* __restrict__ topk_out,     // [B,S,KK]
                                float* __restrict__ wts_out) {  // [B,S,KK]
  const int NTILES = NK_ / 16; // 2048
  int b = blockIdx.y;
  int wave = threadIdx.x >> 5;
  int lane = threadIdx.x & 31;
  int tid  = threadIdx.x;
  int s_base = blockIdx.x * TOKPB_ + wave * 16;

  __shared__ float kt[2][16 * KT_LD];          // 16.5 KB double buffer
  __shared__ float sc_tile[WAVES_][16][16];    // 4 KB
  __shared__ float wf[WAVES_][16][KK_];        // 2 KB
  __shared__ int   ifn[WAVES_][16][KK_];       // 2 KB

  // preload this wave's 16x128 Q tile as 32 A fragments (register resident)
  v2f afrag[32];
  {
    int half = lane >> 4, l = lane & 15;
    const float* qrow = Q + ((long long)b * S_ + s_base + l) * R_;
#pragma unroll
    for (int k = 0; k < 32; ++k) {
      int kk0 = k * 4 + half * 2;
      afrag[k].x = qrow[kk0];
      afrag[k].y = qrow[kk0 + 1];
    }
  }

  float tv[KK_]; int ti[KK_]; // ascending; tv[0] is current min
#pragma unroll
  for (int i = 0; i < KK_; ++i) { tv[i] = -1e30f; ti[i] = 0; }
  const float scale = 0.08838834764831845f; // 1/sqrt(128)

  // prologue: stage tile 0 into buffer 0
  stage_k_tile(Kmat, &kt[0][0], 0, tid);
#if USE_ASYNC_LDS
  __builtin_amdgcn_s_wait_asynccnt(0);
#endif
  __syncthreads();

  for (int nt = 0; nt < NTILES; ++nt) {
    int cur = nt & 1;
    int n_base = nt * 16;
    if (nt + 1 < NTILES)
      stage_k_tile(Kmat, &kt[cur ^ 1][0], (nt + 1) * 16, tid);

    // 16x16x128 tile via 32 chained f32 WMMAs; B fragments from LDS
    v8f acc = {};
    int half = lane >> 4, l = lane & 15;
    const float* ktc = &kt[cur][0];
#pragma unroll
    for (int k = 0; k < 32; ++k) {
      const float* kp = ktc + l * KT_LD + k * 4 + half * 2;
      v2f bf; bf.x = kp[0]; bf.y = kp[1];
      acc = __builtin_amdgcn_wmma_f32_16x16x4_f32(false, afrag[k], false, bf,
                                                  (short)0, acc, false, false);
    }
#pragma unroll
    for (int i = 0; i < 8; ++i)
      sc_tile[wave][half * 8 + i][l] = acc[i] * scale;

    // per-row streaming top-8 (lanes 0..15 each own one row of this wave)
    if (lane < 16) {
      int row = lane;
      for (int c = 0; c < 16; ++c) {
        float v = sc_tile[wave][row][c];
        if (v > tv[0]) {
          int pos = 0;
#pragma unroll
          for (int i = 1; i < KK_; ++i)
            if (v > tv[i]) { tv[i - 1] = tv[i]; ti[i - 1] = ti[i]; pos = i; }
          tv[pos] = v; ti[pos] = n_base + c;
        }
      }
    }

#if USE_ASYNC_LDS
    __builtin_amdgcn_s_wait_asynccnt(0);
#endif
    __syncthreads(); // all waves done with kt[cur]; kt[cur^1] fully staged
  }

  // softmax over the final top-8 and emit weights / indices
  if (lane < 16) {
    int row = lane;
    float m = tv[KK_ - 1], sum = 0.f, w[KK_];
#pragma unroll
    for (int i = 0; i < KK_; ++i) { w[i] = __expf(tv[i] - m); sum += w[i]; }
    float inv = 1.f / sum;
    long long srow = (long long)b * S_ + s_base + row;
#pragma unroll
    for (int i = 0; i < KK_; ++i) { // descending-score order
      float ww = w[KK_ - 1 - i] * inv;
      int id = ti[KK_ - 1 - i];
      wf[wave][row][i] = ww; ifn[wave][row][i] = id;
      wts_out[srow * KK_ + i] = ww;
      topk_out[srow * KK_ + i] = id;
    }
  }
  // same-wave LDS RAW -> ordered by DScnt, no block barrier needed

  // output: each wave gathers V for its own 16 rows (V is L2-resident)
  for (int row = 0; row < 16; ++row) {
    long long srow = (long long)b * S_ + s_base + row;
    for (int d = lane; d < D_; d += 32) {
      float acc = 0.f;
#pragma unroll
      for (int i = 0; i < KK_; ++i)
        acc += wf[wave][row][i] * V[(long long)ifn[wave][row][i] * D_ + d];
      out[srow * D_ + d] = acc;
    }
  }
}

// ---------------------------------------------------------------------------
// Host launcher
// ---------------------------------------------------------------------------
extern "C" void kernel_launch(void* const* d_in, const int* in_sizes, int n_in,
                              void* d_out, int out_size, void* d_ws,
                              size_t ws_size, hipStream_t stream) {
  const float* x        = (const float*)d_in[0]; // [B,S,D]
  const float* Amat     = (const float*)d_in[1]; // [SD,SD]
  const float* Bm       = (const float*)d_in[2]; // [D,SD]
  const float* Wimp     = (const float*)d_in[3]; // [D,SD]
  const float* Wrouter  = (const float*)d_in[4]; // [NC,D]
  const float* compress = (const float*)d_in[5]; // [NC,D,R]
  const float* Kmat     = (const float*)d_in[6]; // [NK,R]
  const float* V        = (const float*)d_in[7]; // [NK,D]

  // d_out layout: output | importance | neuron_w | topk_idx | weights
  float* out      = (float*)d_out;
  float* o_output = out;                                   // B*S*D
  float* o_imp    = out + (long long)B_ * S_ * D_;         // B*S
  float* o_nw     = o_imp + B_ * S_;                       // B*NC
  int*   o_topk   = (int*)(o_nw + B_ * NC_);               // B*S*KK
  float* o_wts    = (float*)(o_topk + B_ * S_ * KK_);      // B*S*KK

  // workspace layout (floats)
  float* ws = (float*)d_ws;
  float* xb     = ws;                                 // B*S*SD = 524288
  float* h_proj = xb + (long long)B_ * S_ * SD_;      // B*D    = 4096
  float* logits = h_proj + B_ * D_;                   // B*S    = 8192
  float* tp     = logits + B_ * S_;                   // B*S*NC = 524288
  float* nw_raw = tp + (long long)B_ * S_ * NC_;      // B*NC   = 256
  float* nw     = nw_raw + B_ * NC_;                  // B*NC   = 256
  float* sc     = nw + B_ * NC_;                      // B*D*R  = 524288
  float* Q      = sc + (long long)B_ * D_ * R_;       // B*S*R  = 1048576

  // 0) zero the neuron accumulator (graph-safe)
  zero_kernel<<<1, 256, 0, stream>>>(nw_raw, B_ * NC_);

  // 1) xB = x @ Bm        [B*S,1024] @ [1024,64]
  wmma_gemm_f32<false><<<dim3((B_ * S_ / 16) * (SD_ / 16)), 32, 0, stream>>>(
      x, Bm, xb, D_, SD_, SD_ / 16, 0, 0, 0);

  // 2) sequential scan + h_proj
  scan_kernel<<<1, 256, 0, stream>>>(xb, Amat, Wimp, h_proj);

  // 3) importance logits + softmax
  imp_logits_kernel<<<B_ * S_, 32, 0, stream>>>(x, h_proj, logits);
  imp_softmax_kernel<<<B_, 256, 0, stream>>>(logits, o_imp);

  // 4) token_pref logits = x @ W_router^T ; softmax ; accumulate neuron_w
  wmma_gemm_f32<true><<<dim3((B_ * S_ / 16) * (NC_ / 16)), 32, 0, stream>>>(
      x, Wrouter, tp, D_, NC_, NC_ / 16, 0, 0, 0);
  tp_accum_kernel<<<B_ * S_ / 256, 256, 0, stream>>>(tp, o_imp, nw_raw);
  nw_norm_kernel<<<B_, NC_, 0, stream>>>(nw_raw, nw, o_nw);

  // 5) shared_compress = neuron_w @ compress
  sc_kernel<<<D_ * R_ / 256, 256, 0, stream>>>(compress, nw, sc);

  // 6) Q = x[b] @ sc[b]   batched WMMA GEMM
  wmma_gemm_f32<false><<<dim3((S_ / 16) * (R_ / 16), B_), 32, 0, stream>>>(
      x, sc, Q, D_, R_, R_ / 16,
      (long long)S_ * D_, (long long)D_ * R_, (long long)S_ * R_);

  // 7) fused scores(WMMA) + top-8 + softmax + weighted V gather,
  //    K tiles async-staged to LDS and shared by 4 waves
  retrieve_kernel<<<dim3(S_ / TOKPB_, B_), WAVES_ * 32, 0, stream>>>(
      Q, Kmat, V, o_output, o_topk, o_wts);
}